// LennardJones_4827543241470
// MI455X (gfx1250) — compile-verified
//
#include <hip/hip_runtime.h>
#include <math.h>

// Lennard-Jones all-pairs energy with minimum-image PBC, tiled for CDNA5
// (gfx1250). B=4, N=4096, D=3 fixed by the reference. Full NxN interaction
// matrix with masked diagonal; final result scaled by 0.5*4*eps = 2*eps.

constexpr int NPART  = 4096;  // particles per batch
constexpr int TI     = 256;   // i-particles per block (1 per thread, 8 wave32s)
constexpr int TJ     = 512;   // j-particles per LDS tile (6 KB AoS)
constexpr int ITILES = NPART / TI;   // 16
constexpr int JTILES = NPART / TJ;   // 8
constexpr int TPB    = ITILES * JTILES;  // 128 tiles per batch (power of two)

// CDNA5 async global->LDS copy: 16 bytes per lane per op, tracked by ASYNCcnt.
__device__ __forceinline__ void async_copy16(unsigned lds_off, const void* gsrc) {
  asm volatile("global_load_async_to_lds_b128 %0, %1, off"
               :: "v"(lds_off), "v"(gsrc)
               : "memory");
}

__device__ __forceinline__ void wait_async0() {
  asm volatile("s_wait_asynccnt 0x0" ::: "memory");
}

__global__ __launch_bounds__(TI) void lj_partial_kernel(
    const float* __restrict__ x,     // (B, N, 3)
    const float* __restrict__ sigp,  // scalar
    const float* __restrict__ boxp,  // (1,)
    float* __restrict__ ws)          // one partial per block
{
  __shared__ float sj[TJ * 3];
  __shared__ float red[TI];

  const int bid = blockIdx.x;
  const int b   = bid >> 7;            // / TPB (128)
  const int rem = bid & (TPB - 1);
  const int it  = rem >> 3;            // / JTILES (8)
  const int jt  = rem & (JTILES - 1);
  const int t   = threadIdx.x;

  const float* xb   = x + (size_t)b * (size_t)NPART * 3;
  const float* jsrc = xb + (size_t)jt * TJ * 3;

  // Stage the j-tile into LDS with CDNA5 async copies.
  // Low 32 bits of the generic pointer to a __shared__ object == LDS byte offset.
  unsigned ldsBase = (unsigned)(size_t)(void*)sj;
  constexpr int CHUNKS = (TJ * 3 * 4) / 16;  // 384 x 16B
  #pragma unroll
  for (int c = t; c < CHUNKS; c += TI) {
    async_copy16(ldsBase + (unsigned)(c * 16), (const char*)jsrc + (size_t)c * 16);
  }

  // Each thread owns one i-particle in registers (overlaps with async copy).
  const int   i  = it * TI + t;
  const float xi = xb[(size_t)i * 3 + 0];
  const float yi = xb[(size_t)i * 3 + 1];
  const float zi = xb[(size_t)i * 3 + 2];
  const float box    = boxp[0];
  const float invbox = 1.0f / box;
  const float sig    = sigp[0];
  const float sig2   = sig * sig;

  wait_async0();       // this wave's async copies landed in LDS
  __syncthreads();     // all waves' copies landed

  float acc = 0.0f;
  int jglob = jt * TJ;
  #pragma unroll 8
  for (int j = 0; j < TJ; ++j, ++jglob) {
    // Uniform LDS address across the wave -> broadcast, conflict-free.
    float dx = xi - sj[j * 3 + 0];
    float dy = yi - sj[j * 3 + 1];
    float dz = zi - sj[j * 3 + 2];
    // minimum image: d - box*round(d/box); rintf == v_rndne_f32 (half-even)
    dx -= box * rintf(dx * invbox);
    dy -= box * rintf(dy * invbox);
    dz -= box * rintf(dz * invbox);
    float r2 = dx * dx + dy * dy + dz * dz;
    // v_rcp_f32 (1 ulp, trans pipe) instead of the ~10-op IEEE div chain;
    // relative error <= ~4e-7 after the 12th power -- below fp32 sum noise.
    float s2 = sig2 * __builtin_amdgcn_rcpf(r2);
    float s6 = s2 * s2 * s2;
    float e  = s6 * s6 - s6;       // NaN on diagonal, masked below
    acc += (i == jglob) ? 0.0f : e;
  }

  // Deterministic fixed-tree block reduction.
  red[t] = acc;
  __syncthreads();
  #pragma unroll
  for (int s = TI / 2; s > 0; s >>= 1) {
    if (t < s) red[t] += red[t + s];
    __syncthreads();
  }
  if (t == 0) ws[bid] = red[0];
}

__global__ __launch_bounds__(TPB) void lj_reduce_kernel(
    const float* __restrict__ ws, const float* __restrict__ epsp,
    float* __restrict__ out)
{
  __shared__ float red[TPB];
  const int b = blockIdx.x;
  const int t = threadIdx.x;
  red[t] = ws[b * TPB + t];
  __syncthreads();
  #pragma unroll
  for (int s = TPB / 2; s > 0; s >>= 1) {
    if (t < s) red[t] += red[t + s];
    __syncthreads();
  }
  // full-matrix sum * 0.5 * 4*eps = 2*eps * sum
  if (t == 0) out[b] = 2.0f * epsp[0] * red[0];
}

extern "C" void kernel_launch(void* const* d_in, const int* in_sizes, int n_in,
                              void* d_out, int out_size, void* d_ws, size_t ws_size,
                              hipStream_t stream) {
  (void)n_in; (void)ws_size; (void)in_sizes;
  const float* x   = (const float*)d_in[0];  // (B,N,3) float32
  const float* eps = (const float*)d_in[1];  // scalar
  const float* sig = (const float*)d_in[2];  // scalar
  const float* box = (const float*)d_in[3];  // (1,)
  float* out = (float*)d_out;
  float* ws  = (float*)d_ws;

  const int B = out_size;            // 4
  const int blocks = B * TPB;        // 512

  lj_partial_kernel<<<blocks, TI, 0, stream>>>(x, sig, box, ws);
  lj_reduce_kernel<<<B, TPB, 0, stream>>>(ws, eps, out);
}